// NGNN_87479893885390
// MI455X (gfx1250) — compile-verified
//
#include <hip/hip_runtime.h>
#include <math.h>

#define N_NODES 100000
#define N_EDGES 1600000
#define IN_F    512
#define OUT_F   256

typedef __attribute__((ext_vector_type(2))) float v2f;
typedef __attribute__((ext_vector_type(8))) float v8f;

// ---------------------------------------------------------------------------
// Kernel 1: C = tanh(A @ W) using V_WMMA_F32_16X16X4_F32 (full f32 precision).
// Block = 256 threads = 8 waves. Block tile: 32 rows x 256 cols (ALL of N),
// so each row stripe of A is read from HBM exactly once (A is 205MB > L2).
// Wave (wm, wn): wm in {0,1} picks 16-row half, wn in {0..3} picks a 64-col
// slab = 4 WMMA N-tiles sharing one A fragment per k-step (4 wmma / A-load).
// Grid = (100000/32, 1). K = 512 -> 4x128 = 512 WMMA per wave.
// ---------------------------------------------------------------------------
__global__ __launch_bounds__(256) void ngnn_gemm_tanh_wmma(
    const float* __restrict__ A,   // [N_NODES, IN_F] row-major
    const float* __restrict__ W,   // [IN_F, OUT_F]  row-major
    float* __restrict__ C)         // [N_NODES, OUT_F]
{
    const int lane = threadIdx.x & 31;
    const int wave = threadIdx.x >> 5;
    const int wm   = wave >> 2;          // 0..1  (16-row half of block tile)
    const int wn   = wave & 3;           // 0..3  (64-col slab)
    const int m0   = blockIdx.x * 32 + wm * 16;
    const int n0   = wn * 64;

    const int half = lane >> 4;          // 0: K offs {0,1}, 1: K offs {2,3}
    const int l16  = lane & 15;

    // A fragment: lane l16 holds row (m0+l16), K = k + half*2 + {0,1} (contiguous)
    const float* aptr = A + (size_t)(m0 + l16) * IN_F + half * 2;
    // B fragments: lane l16 holds col (n0+l16+16*t), K rows k + half*2 + {0,1}
    const float* bptr = W + (size_t)(half * 2) * OUT_F + (n0 + l16);

    v8f acc0 = {0.f,0.f,0.f,0.f,0.f,0.f,0.f,0.f};
    v8f acc1 = acc0, acc2 = acc0, acc3 = acc0;

#pragma unroll 2
    for (int k = 0; k < IN_F; k += 4) {
        const v2f a = *(const v2f*)(aptr + k);        // one A frag, 4 wmma uses
        const float* bk0 = bptr + (size_t)k * OUT_F;
        const float* bk1 = bptr + (size_t)(k + 1) * OUT_F;
        v2f b0, b1, b2, b3;
        b0.x = bk0[0];  b0.y = bk1[0];
        b1.x = bk0[16]; b1.y = bk1[16];
        b2.x = bk0[32]; b2.y = bk1[32];
        b3.x = bk0[48]; b3.y = bk1[48];
        // (neg_a, A, neg_b, B, c_mod, C, reuse_a, reuse_b)
        acc0 = __builtin_amdgcn_wmma_f32_16x16x4_f32(false, a, false, b0, (short)0, acc0, false, false);
        acc1 = __builtin_amdgcn_wmma_f32_16x16x4_f32(false, a, false, b1, (short)0, acc1, false, false);
        acc2 = __builtin_amdgcn_wmma_f32_16x16x4_f32(false, a, false, b2, (short)0, acc2, false, false);
        acc3 = __builtin_amdgcn_wmma_f32_16x16x4_f32(false, a, false, b3, (short)0, acc3, false, false);
    }

    // D layout: VGPR r -> M = m0 + r + half*8, N = n0 + 16*t + l16
    float* cptr = C + (size_t)(m0 + half * 8) * OUT_F + n0 + l16;
#pragma unroll
    for (int r = 0; r < 8; ++r) {
        float* cr = cptr + (size_t)r * OUT_F;
        cr[0]  = tanhf(acc0[r]);
        cr[16] = tanhf(acc1[r]);
        cr[32] = tanhf(acc2[r]);
        cr[48] = tanhf(acc3[r]);
    }
}

// ---------------------------------------------------------------------------
// Kernel 2: CSR pointer build from sorted COO row array (lower_bound per node).
// ---------------------------------------------------------------------------
__global__ __launch_bounds__(256) void ngnn_build_row_ptr(
    const int* __restrict__ row, int* __restrict__ row_ptr)
{
    const int i = blockIdx.x * blockDim.x + threadIdx.x;
    if (i > N_NODES) return;
    int lo = 0, hi = N_EDGES;                 // first e with row[e] >= i
    while (lo < hi) {
        const int mid = (lo + hi) >> 1;
        if (row[mid] < i) lo = mid + 1; else hi = mid;
    }
    row_ptr[i] = lo;
}

// ---------------------------------------------------------------------------
// Kernel 3: SpMM, one wave per output row. Each lane owns 8 of 256 features
// (2x float4). Gathers x[col[e]] are coalesced 1KB row reads (L2-resident:
// x is 102.4 MB < 192 MB L2). Registers accumulate; single coalesced store.
// ---------------------------------------------------------------------------
__global__ __launch_bounds__(256) void ngnn_spmm_row_wave(
    const int* __restrict__ row_ptr,
    const int* __restrict__ col,
    const float* __restrict__ vals,
    const float* __restrict__ x,   // [N_NODES, OUT_F]
    float* __restrict__ y)         // [N_NODES, OUT_F]
{
    const int lane = threadIdx.x & 31;
    const int wave = threadIdx.x >> 5;
    const int node = blockIdx.x * 8 + wave;
    if (node >= N_NODES) return;

    const int e0 = row_ptr[node];
    const int e1 = row_ptr[node + 1];

    float4 acc0 = make_float4(0.f, 0.f, 0.f, 0.f);
    float4 acc1 = make_float4(0.f, 0.f, 0.f, 0.f);

    for (int e = e0; e < e1; ++e) {
        const int   c = col[e];
        const float v = vals[e];
        const float* xr = x + (size_t)c * OUT_F;
        if (e + 1 < e1) {  // hide L2 gather latency for the next edge's row
            __builtin_prefetch(x + (size_t)col[e + 1] * OUT_F + lane * 4, 0, 3);
        }
        const float4 a = *(const float4*)(xr + lane * 4);
        const float4 b = *(const float4*)(xr + 128 + lane * 4);
        acc0.x = fmaf(v, a.x, acc0.x); acc0.y = fmaf(v, a.y, acc0.y);
        acc0.z = fmaf(v, a.z, acc0.z); acc0.w = fmaf(v, a.w, acc0.w);
        acc1.x = fmaf(v, b.x, acc1.x); acc1.y = fmaf(v, b.y, acc1.y);
        acc1.z = fmaf(v, b.z, acc1.z); acc1.w = fmaf(v, b.w, acc1.w);
    }

    float* yr = y + (size_t)node * OUT_F;
    *(float4*)(yr + lane * 4)       = acc0;
    *(float4*)(yr + 128 + lane * 4) = acc1;
}

// ---------------------------------------------------------------------------
// Launcher. inputs: features f32, weight f32, row i32, col i32, adj_vals f32.
// ws layout: [row_ptr: (N_NODES+1) ints, 256B padded][tmp: N_NODES*OUT_F f32]
// Pipeline: gemm -> d_out ; spmm(d_out)->tmp ; spmm(tmp)->d_out.
// ---------------------------------------------------------------------------
extern "C" void kernel_launch(void* const* d_in, const int* in_sizes, int n_in,
                              void* d_out, int out_size, void* d_ws, size_t ws_size,
                              hipStream_t stream) {
    const float* features = (const float*)d_in[0];
    const float* weight   = (const float*)d_in[1];
    const int*   row      = (const int*)d_in[2];
    const int*   col      = (const int*)d_in[3];
    const float* adj_vals = (const float*)d_in[4];
    float*       out      = (float*)d_out;

    int*   row_ptr  = (int*)d_ws;
    size_t rp_bytes = (((size_t)(N_NODES + 1) * sizeof(int)) + 255) & ~(size_t)255;
    float* tmp      = (float*)((char*)d_ws + rp_bytes);

    // CSR pointers (independent of GEMM; ordered on the stream)
    {
        dim3 grid((N_NODES + 1 + 255) / 256), block(256);
        ngnn_build_row_ptr<<<grid, block, 0, stream>>>(row, row_ptr);
    }
    // tanh(features @ weight) -> out   (A read exactly once from HBM)
    {
        dim3 grid(N_NODES / 32, 1), block(256);
        ngnn_gemm_tanh_wmma<<<grid, block, 0, stream>>>(features, weight, out);
    }
    // spmm pass 1: out -> tmp
    {
        dim3 grid((N_NODES + 7) / 8), block(256);
        ngnn_spmm_row_wave<<<grid, block, 0, stream>>>(row_ptr, col, adj_vals, out, tmp);
    }
    // spmm pass 2: tmp -> out
    {
        dim3 grid((N_NODES + 7) / 8), block(256);
        ngnn_spmm_row_wave<<<grid, block, 0, stream>>>(row_ptr, col, adj_vals, tmp, out);
    }
}